// Sinkhorn_68332929679872
// MI455X (gfx1250) — compile-verified
//
#include <hip/hip_runtime.h>
#include <math.h>

// Sinkhorn (log-space) collapsed to dual-vector form:
//   log_s = A - R_i - C_j  at all times, A = s / tau.
//   even it: R_i = LSE_j(A - C_j);  odd it: C_j = LSE_i(A - R_i)
//   out = exp(A - R - C)
// Fused row+col pass: each 4-row panel is read from HBM ONCE per (row,col)
// iteration pair. Panels double-buffered in 256KB of CDNA5's 320KB WGP LDS,
// streamed by the async global->LDS engine (ASYNCcnt) to overlap load+compute.
// Traffic: 5 matrix reads + partials + final read/write (~2.0 GB @ 23.3 TB/s).

#define N        8192
#define TPB      256      // small kernels
#define TPBF     1024     // fused kernel (32 waves)
#define PR       4        // rows per panel (128 KB per buffer)
#define BLOCKS_F 256
#define PANELS_PER_BLOCK (N / (PR * BLOCKS_F))   // 8
#define NSEGF    BLOCKS_F                        // 256 column-partial segments
#define INV_TAU  1.0f                            // TAU = 1.0

// Static device scratch: rewritten fully each call -> deterministic.
__device__ __align__(16) float g_R[N];
__device__ __align__(16) float g_C[N];
__device__ __align__(16) float g_pm[NSEGF][N];
__device__ __align__(16) float g_ps[NSEGF][N];

// ---------------- CDNA5 async global->LDS path ----------------
#if defined(__HIP_DEVICE_COMPILE__) && defined(__has_builtin)
#if __has_builtin(__builtin_amdgcn_global_load_async_to_lds_b128)
#define HAVE_ASYNC_LDS 1
#endif
#endif

#ifdef HAVE_ASYNC_LDS
typedef int v4i_t __attribute__((ext_vector_type(4)));
typedef __attribute__((address_space(1))) v4i_t* as1_v4i;
typedef __attribute__((address_space(3))) v4i_t* as3_v4i;
#define ASYNC_LOAD16(gp, lp) \
  __builtin_amdgcn_global_load_async_to_lds_b128((as1_v4i)(gp), (as3_v4i)(lp), 0, 0)
#if __has_builtin(__builtin_amdgcn_s_wait_asynccnt)
#define WAIT_ASYNC(n) __builtin_amdgcn_s_wait_asynccnt(n)
#else
#define WAIT_ASYNC(n) asm volatile("s_wait_asynccnt %0" ::"i"(n))
#endif
#endif

// ---------------- kernels ----------------

__global__ void __launch_bounds__(TPB) zero_C_kernel() {
  g_C[blockIdx.x * TPB + threadIdx.x] = 0.0f;
}

// One fused pass = one row iteration + (read side of) one column iteration.
// Block: 1024 threads, loops over 8 panels of 4 rows x 8192 cols,
// double-buffered via async global->LDS (8 async instrs/wave per panel).
__global__ void __launch_bounds__(TPBF) fused_row_col_kernel(const float* __restrict__ s) {
  __shared__ __align__(16) float panel[2 * PR * N];   // 2 x 128 KB
  __shared__ float wred[PR * 8];                      // 4 rows x 8 waves
  __shared__ float Rsh[PR];

  const int t    = threadIdx.x;
  const int lane = t & 31;
  const int r    = t >> 8;          // row group (256 threads per row)
  const int c    = t & 255;         // thread index within row group
  const int wir  = (t >> 5) & 7;    // wave index within row group

  float m_c[8], s_c[8];             // per-thread column partials (8 cols)
#pragma unroll
  for (int k = 0; k < 8; ++k) { m_c[k] = -INFINITY; s_c[k] = 0.0f; }

#ifdef HAVE_ASYNC_LDS
  auto issue = [&](float* dst, int pidx) {
    const int row0 = (blockIdx.x * PANELS_PER_BLOCK + pidx) * PR;
#pragma unroll
    for (int u = 0; u < 8; ++u) {
      const int q = t + u * TPBF;                     // float4 index in panel
      const float* gp = s + (size_t)row0 * N + (size_t)q * 4;
      ASYNC_LOAD16(gp, dst + q * 4);
    }
  };
  issue(&panel[0], 0);
#endif

  for (int p = 0; p < PANELS_PER_BLOCK; ++p) {
    const int cb   = p & 1;
    const int row0 = (blockIdx.x * PANELS_PER_BLOCK + p) * PR;
    float* pb = &panel[(size_t)cb * (PR * N)];

#ifdef HAVE_ASYNC_LDS
    if (p + 1 < PANELS_PER_BLOCK) {
      issue(&panel[(size_t)(cb ^ 1) * (PR * N)], p + 1);
      WAIT_ASYNC(8);                // current panel done; next 8 in flight
    } else {
      WAIT_ASYNC(0);
    }
#else
#pragma unroll
    for (int u = 0; u < 8; ++u) {
      const int q = t + u * TPBF;
      ((float4*)pb)[q] = ((const float4*)(s + (size_t)row0 * N))[q];
    }
#endif
    __syncthreads();

    // ---- phase 1: row LSE  R_i = LSE_j(A - C_j) ----
    float mloc = -INFINITY;
#pragma unroll 8
    for (int e = 0; e < 32; ++e) {
      const int col = c + e * 256;                    // conflict-free LDS
      const float x = pb[r * N + col] * INV_TAU - g_C[col];
      mloc = fmaxf(mloc, x);
    }
#pragma unroll
    for (int off = 16; off > 0; off >>= 1)
      mloc = fmaxf(mloc, __shfl_xor(mloc, off));
    if (lane == 0) wred[r * 8 + wir] = mloc;
    __syncthreads();
    float M = wred[r * 8 + 0];
#pragma unroll
    for (int k = 1; k < 8; ++k) M = fmaxf(M, wred[r * 8 + k]);
    __syncthreads();                                  // before wred reuse

    float sloc = 0.0f;
#pragma unroll 8
    for (int e = 0; e < 32; ++e) {
      const int col = c + e * 256;
      const float x = pb[r * N + col] * INV_TAU - g_C[col];
      sloc += __expf(x - M);
    }
#pragma unroll
    for (int off = 16; off > 0; off >>= 1)
      sloc += __shfl_xor(sloc, off);
    if (lane == 0) wred[r * 8 + wir] = sloc;
    __syncthreads();
    if (c == 0) {
      float ssum = 0.0f;
#pragma unroll
      for (int k = 0; k < 8; ++k) ssum += wred[r * 8 + k];
      const float Rv = M + __logf(ssum);
      Rsh[r] = Rv;
      g_R[row0 + r] = Rv;
    }
    __syncthreads();

    // ---- phase 2: column partials with the NEW R (two-level LSE merge) ----
#pragma unroll
    for (int k = 0; k < 8; ++k) {
      const int j = t + k * TPBF;
      float x0 = pb[0 * N + j] * INV_TAU - Rsh[0];
      float x1 = pb[1 * N + j] * INV_TAU - Rsh[1];
      float x2 = pb[2 * N + j] * INV_TAU - Rsh[2];
      float x3 = pb[3 * N + j] * INV_TAU - Rsh[3];
      const float pm = fmaxf(fmaxf(x0, x1), fmaxf(x2, x3));
      const float ps = __expf(x0 - pm) + __expf(x1 - pm) +
                       __expf(x2 - pm) + __expf(x3 - pm);
      const float nm = fmaxf(m_c[k], pm);
      s_c[k] = s_c[k] * __expf(m_c[k] - nm) + ps * __expf(pm - nm);
      m_c[k] = nm;
    }
    __syncthreads();                                  // buffer reused/refilled
  }

  // ---- write merged column partials (one segment per block) ----
#pragma unroll
  for (int k = 0; k < 8; ++k) {
    const int j = t + k * TPBF;
    g_pm[blockIdx.x][j] = m_c[k];
    g_ps[blockIdx.x][j] = s_c[k];
  }
}

// Merge NSEGF per-block (max,sum) partials -> C_j
__global__ void __launch_bounds__(TPB) col_combine_kernel() {
  const int j = blockIdx.x * TPB + threadIdx.x;
  float m = -INFINITY;
  for (int k = 0; k < NSEGF; ++k) m = fmaxf(m, g_pm[k][j]);
  float acc = 0.0f;
  for (int k = 0; k < NSEGF; ++k) acc += g_ps[k][j] * __expf(g_pm[k][j] - m);
  g_C[j] = m + __logf(acc);
}

// out[i,j] = exp(s*invTau - R_i - C_j)
__global__ void __launch_bounds__(TPB) finalize_kernel(const float* __restrict__ s,
                                                       float* __restrict__ out) {
  const int t = threadIdx.x;
  const int i = blockIdx.y;
  const int q = blockIdx.x * TPB + t;                 // float4 index within row
  const float4 v = ((const float4*)(s + (size_t)i * N))[q];
  const float4 c = ((const float4*)g_C)[q];
  const float Ri = g_R[i];
  float4 o;
  o.x = __expf(v.x * INV_TAU - Ri - c.x);
  o.y = __expf(v.y * INV_TAU - Ri - c.y);
  o.z = __expf(v.z * INV_TAU - Ri - c.z);
  o.w = __expf(v.w * INV_TAU - Ri - c.w);
  ((float4*)(out + (size_t)i * N))[q] = o;
}

// ---------------- launch ----------------

extern "C" void kernel_launch(void* const* d_in, const int* in_sizes, int n_in,
                              void* d_out, int out_size, void* d_ws, size_t ws_size,
                              hipStream_t stream) {
  (void)in_sizes; (void)n_in; (void)out_size; (void)d_ws; (void)ws_size;
  const float* s = (const float*)d_in[0];
  float* out = (float*)d_out;

  zero_C_kernel<<<N / TPB, TPB, 0, stream>>>();

  // 10 iterations = 5 fused (row it 2k, col it 2k+1) passes.
  for (int pass = 0; pass < 5; ++pass) {
    fused_row_col_kernel<<<BLOCKS_F, TPBF, 0, stream>>>(s);
    col_combine_kernel<<<N / TPB, TPB, 0, stream>>>();
  }

  finalize_kernel<<<dim3(N / (TPB * 4), N), TPB, 0, stream>>>(s, out);
}